// GCN_27101243638200
// MI455X (gfx1250) — compile-verified
//
#include <hip/hip_runtime.h>
#include <hip/hip_bf16.h>

typedef __attribute__((ext_vector_type(2))) float v2f;
typedef __attribute__((ext_vector_type(8))) float v8f;

#define IN_FEATS  128
#define OUT_FEATS 128

// ---------------------------------------------------------------------------
// Phase 0: zero the aggregation accumulator (d_ws), 16B per thread.
// ---------------------------------------------------------------------------
__global__ __launch_bounds__(256) void gcn_zero_kernel(float4* __restrict__ p, int n4) {
    int i = blockIdx.x * blockDim.x + threadIdx.x;
    if (i < n4) p[i] = make_float4(0.f, 0.f, 0.f, 0.f);
}

// ---------------------------------------------------------------------------
// Phase 1: COO SpMM scatter-add. One wave32 per edge; lane l covers feature
// elements 4l..4l+3 (32 lanes * 4 = 128 feats). Uses hardware fp32 global
// atomics (global_atomic_add_f32) which execute in L2's atomic units.
// ---------------------------------------------------------------------------
__global__ __launch_bounds__(256) void gcn_spmm_kernel(
        const int*   __restrict__ erow,
        const int*   __restrict__ ecol,
        const float* __restrict__ eval,
        const float* __restrict__ X,
        float*       __restrict__ agg,
        int E) {
    const int wave = (blockIdx.x * 256 + threadIdx.x) >> 5;
    const int lane = threadIdx.x & 31;
    if (wave >= E) return;

    const int   r = erow[wave];
    const int   c = ecol[wave];
    const float v = eval[wave];

    const float4 x = ((const float4*)(X + (size_t)c * IN_FEATS))[lane];
    float* dst = agg + (size_t)r * IN_FEATS + lane * 4;
    unsafeAtomicAdd(dst + 0, x.x * v);
    unsafeAtomicAdd(dst + 1, x.y * v);
    unsafeAtomicAdd(dst + 2, x.z * v);
    unsafeAtomicAdd(dst + 3, x.w * v);
}

// ---------------------------------------------------------------------------
// Phase 2: dense GEMM (agg[M,128] @ W[128,128]) + ReLU, fp32 WMMA.
//
// W is staged in LDS in a K-PAIR-INTERLEAVED layout:
//     Wlds[(k>>1)*256 + n*2 + (k&1)] = W[k][n]
// so a B fragment {W[k][n], W[k+1][n]} is one aligned 8-byte LDS word ->
// a single ds_load_b64 straight into the even-aligned VGPR pair that
// V_WMMA_F32_16X16X4_F32 consumes (no repacking movs).
//
// A fragment (16x4, 2 VGPRs): lanes 0-15 hold K=0,1; lanes 16-31 hold K=2,3.
// B fragment (4x16, 2 VGPRs): lanes 0-15 hold K=0,1 for N=lane; 16-31 K=2,3.
// C/D (16x16, 8 VGPRs): VGPR r = row r (lanes 0-15) / row r+8 (lanes 16-31).
// ---------------------------------------------------------------------------
__global__ __launch_bounds__(256) void gcn_gemm_relu_kernel(
        const float* __restrict__ agg,
        const float* __restrict__ W,
        float*       __restrict__ out,
        int M) {
    __shared__ float Wlds[IN_FEATS * OUT_FEATS];   // 64 KB, pair-interleaved

    // Cooperative load + interleave of the weight matrix into LDS.
    // Global reads stay fully coalesced (consecutive i -> consecutive addr).
    #pragma unroll 4
    for (int i = threadIdx.x; i < IN_FEATS * OUT_FEATS; i += 256) {
        const int k = i >> 7;        // row (K index)
        const int n = i & 127;       // col (N index)
        Wlds[(k >> 1) * 256 + n * 2 + (k & 1)] = W[i];
    }
    __syncthreads();

    const int wave = threadIdx.x >> 5;
    const int lane = threadIdx.x & 31;
    const int tile = blockIdx.x * 8 + wave;
    const int m0   = tile * 16;
    if (m0 >= M) return;                 // wave-uniform: EXEC stays all-ones

    const int half = lane >> 4;          // 0: K sub 0,1   1: K sub 2,3
    const int l16  = lane & 15;
    const int koff = half * 2;           // even -> 8B-aligned fragment address

    const float* arow = agg + (size_t)(m0 + l16) * IN_FEATS;

    v8f acc[8] = {};                     // 8 N-tiles of 16x16 f32

    #pragma unroll 8
    for (int k = 0; k < IN_FEATS; k += 4) {
        v2f a;
        const float* ap = arow + k + koff;
        a.x = ap[0];                     // A[m][k+koff]     (merges to b64 load)
        a.y = ap[1];                     // A[m][k+koff+1]

        // Base of the interleaved K-pair row (k+koff is even).
        const float* bbase = Wlds + ((k + koff) >> 1) * 256 + l16 * 2;

        #pragma unroll
        for (int n = 0; n < 8; ++n) {
            // {W[k+koff][n16], W[k+koff+1][n16]} : one ds_load_b64.
            v2f b = *(const v2f*)(bbase + n * 32);
            acc[n] = __builtin_amdgcn_wmma_f32_16x16x4_f32(
                         false, a, false, b, (short)0, acc[n], false, false);
        }
    }

    // Fused ReLU + store. Lane layout: col = n*16 + l16, row = m0 + r + 8*half.
    #pragma unroll
    for (int n = 0; n < 8; ++n) {
        const int col = n * 16 + l16;
        #pragma unroll
        for (int r = 0; r < 8; ++r) {
            const int row = m0 + r + half * 8;
            float v = acc[n][r];
            out[(size_t)row * OUT_FEATS + col] = v > 0.f ? v : 0.f;
        }
    }
}

// ---------------------------------------------------------------------------
extern "C" void kernel_launch(void* const* d_in, const int* in_sizes, int n_in,
                              void* d_out, int out_size, void* d_ws, size_t ws_size,
                              hipStream_t stream) {
    const int*   edge_row  = (const int*)  d_in[0];
    const int*   edge_col  = (const int*)  d_in[1];
    const float* edge_vals = (const float*)d_in[2];
    const float* node_embs = (const float*)d_in[3];
    const float* gcn_w     = (const float*)d_in[4];
    float*       out       = (float*)d_out;
    float*       agg       = (float*)d_ws;         // [N, 128] fp32 accumulator

    const int E = in_sizes[0];                     // 600000 edges
    const int N = in_sizes[3] / IN_FEATS;          // 50000 nodes

    // Phase 0: zero accumulator (ws is poisoned / stale between calls).
    {
        const int n4 = (N * IN_FEATS) / 4;
        gcn_zero_kernel<<<(n4 + 255) / 256, 256, 0, stream>>>((float4*)agg, n4);
    }

    // Phase 1: edge scatter-add (8 waves = 8 edges per 256-thread block).
    gcn_spmm_kernel<<<(E + 7) / 8, 256, 0, stream>>>(
        edge_row, edge_col, edge_vals, node_embs, agg, E);

    // Phase 2: WMMA GEMM + ReLU. 16-row tiles, 8 tiles per block.
    {
        const int tiles  = (N + 15) / 16;          // 3125
        const int blocks = (tiles + 7) / 8;        // 391
        gcn_gemm_relu_kernel<<<blocks, 256, 0, stream>>>(agg, gcn_w, out, N);
    }
}